// GraphConvLSTM_10763188044359
// MI455X (gfx1250) — compile-verified
//
#include <hip/hip_runtime.h>
#include <math.h>

#define SEQ_LEN 16
#define IN_SIZE 32
#define HID     64
#define N_NODES 20000
#define N_EDGES 320000
#define BATCH   4
#define BNH     ((size_t)BATCH * N_NODES * HID)   // 5,120,000 floats

typedef __attribute__((ext_vector_type(2))) float v2f;
typedef __attribute__((ext_vector_type(8))) float v8f;

__device__ __forceinline__ float sigmoidf_fast(float x) {
    return 1.0f / (1.0f + __expf(-x));
}

// ---------------------------------------------------------------- degrees
__global__ void degree_kernel(const int* __restrict__ src,
                              const int* __restrict__ dst,
                              float* __restrict__ deg_out,
                              float* __restrict__ deg_in) {
    int e = blockIdx.x * blockDim.x + threadIdx.x;   // grid exact: E/256
    atomicAdd(&deg_out[src[e]], 1.0f);
    atomicAdd(&deg_in[dst[e]], 1.0f);
}

__global__ void norm_kernel(float* __restrict__ ns, float* __restrict__ nd) {
    int i = blockIdx.x * blockDim.x + threadIdx.x;
    if (i < N_NODES) {
        ns[i] = rsqrtf(fmaxf(ns[i], 1.0f));
        nd[i] = rsqrtf(fmaxf(nd[i], 1.0f));
    }
}

// ------------------------------------------------- x gate projections (tiny)
__global__ void xproj_kernel(const float* __restrict__ x,
                             const float* __restrict__ Wi, const float* __restrict__ bi,
                             const float* __restrict__ Wf, const float* __restrict__ bf,
                             const float* __restrict__ Wo, const float* __restrict__ bo,
                             const float* __restrict__ Wc, const float* __restrict__ bc,
                             float* __restrict__ xi, float* __restrict__ xf,
                             float* __restrict__ xo, float* __restrict__ xc) {
    int tid = blockIdx.x * blockDim.x + threadIdx.x;
    if (tid >= BATCH * HID) return;
    int b = tid / HID, h = tid % HID;
    float si = bi[h], sf = bf[h], so = bo[h], sc = bc[h];
#pragma unroll
    for (int k = 0; k < IN_SIZE; ++k) {
        float xv = x[b * IN_SIZE + k];
        si += xv * Wi[k * HID + h];
        sf += xv * Wf[k * HID + h];
        so += xv * Wo[k * HID + h];
        sc += xv * Wc[k * HID + h];
    }
    xi[tid] = si; xf[tid] = sf; xo[tid] = so; xc[tid] = sc;
}

// ------------------------------------- f = (h * norm_src) @ Wg  via f32 WMMA
// grid: (125, BATCH), block: 320 (10 waves). 125*10 = 1250 tiles = 20000 rows
// exactly -> no guards, no divergence, straight-line load/WMMA stream.
__global__ __launch_bounds__(320) void gcn_gemm_wmma(
        const float* __restrict__ hbuf, const float* __restrict__ ns,
        const float* __restrict__ Wg, float* __restrict__ f) {
    // Pre-swizzled Wg: wswz[kg][col][half][j] = Wg[kg*4 + half*2 + j][col]
    // -> each lane's B fragment (K, K+1) is one contiguous ds_load_b64,
    //    and the 32-lane access covers all 64 LDS banks exactly once.
    __shared__ float wswz[HID / 4 * HID * 2 * 2];   // 4096 floats = 16 KB
    for (int i = threadIdx.x; i < HID * HID; i += blockDim.x) {
        int k    = i >> 6;          // Wg row 0..63
        int col  = i & 63;
        int kg   = k >> 2;
        int half = (k >> 1) & 1;
        int j    = k & 1;
        wswz[((kg * HID + col) * 2 + half) * 2 + j] = Wg[i];
    }
    __syncthreads();

    const int wave = threadIdx.x >> 5;
    const int lane = threadIdx.x & 31;
    const int half = lane >> 4;                 // 0: lanes 0-15, 1: lanes 16-31
    const int l15  = lane & 15;
    const int tile = blockIdx.x * 10 + wave;    // 16-row tile index (< 1250)
    const int b    = blockIdx.y;
    const int row_base = tile * 16;

    const int   arow = row_base + l15;          // always < N_NODES
    const float nsv  = ns[arow];
    const float2* arowp = (const float2*)(hbuf + ((size_t)b * N_NODES + arow) * HID);

    v8f acc[4];
#pragma unroll
    for (int ct = 0; ct < 4; ++ct)
#pragma unroll
        for (int r = 0; r < 8; ++r) acc[ct][r] = 0.0f;

#pragma unroll
    for (int kb = 0; kb < HID; kb += 4) {
        const int kg = kb >> 2;
        // A 16x4 f32 fragment: lanes 0-15 K={kb,kb+1}, lanes 16-31 K={kb+2,kb+3}
        const float2 h2 = arowp[(kb >> 1) + half];
        v2f a; a.x = h2.x * nsv; a.y = h2.y * nsv;
#pragma unroll
        for (int ct = 0; ct < 4; ++ct) {
            const int col = ct * 16 + l15;
            const float2 w2 = *(const float2*)&wswz[((kg * HID + col) * 2 + half) * 2];
            v2f bb; bb.x = w2.x; bb.y = w2.y;
            acc[ct] = __builtin_amdgcn_wmma_f32_16x16x4_f32(
                false, a, false, bb, (short)0, acc[ct], false, false);
        }
    }

    float* fb = f + (size_t)b * N_NODES * HID;
#pragma unroll
    for (int ct = 0; ct < 4; ++ct) {
        const int col = ct * 16 + l15;
#pragma unroll
        for (int r = 0; r < 8; ++r) {
            const int row = row_base + r + half * 8;   // C vgpr r -> M=r / M=8+r
            fb[(size_t)row * HID + col] = acc[ct][r];
        }
    }
}

// ------------------------------------------- agg[b][dst] += f[b][src] (E,B,H)
// grid: (E*16/256 = 20000, BATCH); 16 threads cooperate per edge, float4 each
__global__ __launch_bounds__(256) void scatter_kernel(
        const int* __restrict__ src, const int* __restrict__ dst,
        const float* __restrict__ f, float* __restrict__ agg) {
    const unsigned tid = blockIdx.x * blockDim.x + threadIdx.x;
    const unsigned e = tid >> 4;       // edge index (grid exact: E*16/256)
    const unsigned j = tid & 15;       // which float4 of the 64-float row
    const int b = blockIdx.y;
    const int s = src[e];
    const int d = dst[e];
    const float4* fp = (const float4*)(f + ((size_t)b * N_NODES + s) * HID) + j;
    const float4 v = *fp;
    float* ap = agg + ((size_t)b * N_NODES + d) * HID + j * 4;
    atomicAdd(ap + 0, v.x);
    atomicAdd(ap + 1, v.y);
    atomicAdd(ap + 2, v.z);
    atomicAdd(ap + 3, v.w);
}

// ------------------------------ fused: gh, 4 gates, LSTM update, output write
// grid: (N_NODES*HID/256 = 5000, BATCH), exact
__global__ __launch_bounds__(256) void gate_kernel(
        const float* __restrict__ agg, const float* __restrict__ nd,
        const float* __restrict__ bg,
        const float* __restrict__ xi, const float* __restrict__ xf,
        const float* __restrict__ xo, const float* __restrict__ xc,
        float* __restrict__ hbuf, float* __restrict__ cbuf,
        float* __restrict__ out, int t) {
    const size_t i = (size_t)blockIdx.x * blockDim.x + threadIdx.x; // n*HID+h
    const int b  = blockIdx.y;
    const int hh = (int)(i & (HID - 1));
    const int n  = (int)(i >> 6);
    const size_t idx = (size_t)b * N_NODES * HID + i;

    const float gh = agg[idx] * nd[n] + bg[hh];
    const int g = b * HID + hh;
    const float i_t = sigmoidf_fast(xi[g] + gh);
    const float f_t = sigmoidf_fast(xf[g] + gh);
    const float o_t = sigmoidf_fast(xo[g] + gh);
    const float c_t = tanhf(xc[g] + gh);
    const float c_new = f_t * cbuf[idx] + i_t * c_t;
    const float h_new = o_t * tanhf(c_new);
    cbuf[idx] = c_new;
    hbuf[idx] = h_new;
    // out[b][t][n][h]
    out[(((size_t)b * SEQ_LEN + t) * N_NODES + n) * HID + hh] = h_new;
}

// ---------------------------------------------------------------------------
extern "C" void kernel_launch(void* const* d_in, const int* in_sizes, int n_in,
                              void* d_out, int out_size, void* d_ws, size_t ws_size,
                              hipStream_t stream) {
    const float* x   = (const float*)d_in[0];
    const float* Wi  = (const float*)d_in[1];
    const float* bi  = (const float*)d_in[2];
    const float* Wf  = (const float*)d_in[3];
    const float* bf  = (const float*)d_in[4];
    const float* Wo  = (const float*)d_in[5];
    const float* bo  = (const float*)d_in[6];
    const float* Wc  = (const float*)d_in[7];
    const float* bc  = (const float*)d_in[8];
    const float* Wg  = (const float*)d_in[9];
    const float* bg  = (const float*)d_in[10];
    const int*   src = (const int*)d_in[11];
    const int*   dst = (const int*)d_in[12];
    float* out = (float*)d_out;

    float* ws   = (float*)d_ws;
    float* hbuf = ws;                   // BNH
    float* cbuf = hbuf + BNH;           // BNH
    float* fbuf = cbuf + BNH;           // BNH
    float* agg  = fbuf + BNH;           // BNH
    float* ns   = agg + BNH;            // N_NODES
    float* nd   = ns + N_NODES;         // N_NODES
    float* xi   = nd + N_NODES;         // BATCH*HID each
    float* xf   = xi + BATCH * HID;
    float* xo   = xf + BATCH * HID;
    float* xc   = xo + BATCH * HID;

    // degrees -> norms (ns/nd double as accumulators)
    hipMemsetAsync(ns, 0, 2 * N_NODES * sizeof(float), stream);
    degree_kernel<<<N_EDGES / 256, 256, 0, stream>>>(src, dst, ns, nd);
    norm_kernel<<<(N_NODES + 255) / 256, 256, 0, stream>>>(ns, nd);

    // x projections (fixed across timesteps)
    xproj_kernel<<<1, 256, 0, stream>>>(x, Wi, bi, Wf, bf, Wo, bo, Wc, bc,
                                        xi, xf, xo, xc);

    // h0 = c0 = 0 (contiguous)
    hipMemsetAsync(hbuf, 0, 2 * BNH * sizeof(float), stream);

    const dim3 gemm_grid(125, BATCH);                 // 125 blocks * 10 waves = 1250 tiles
    const dim3 scat_grid(N_EDGES * 16 / 256, BATCH);  // 20000 blocks per batch
    const dim3 gate_grid(N_NODES * HID / 256, BATCH); // 5000 blocks, exact

    for (int t = 0; t < SEQ_LEN; ++t) {
        gcn_gemm_wmma<<<gemm_grid, 320, 0, stream>>>(hbuf, ns, Wg, fbuf);
        hipMemsetAsync(agg, 0, BNH * sizeof(float), stream);
        scatter_kernel<<<scat_grid, 256, 0, stream>>>(src, dst, fbuf, agg);
        gate_kernel<<<gate_grid, 256, 0, stream>>>(agg, nd, bg, xi, xf, xo, xc,
                                                   hbuf, cbuf, out, t);
    }
}